// CompLayer_50448685859250
// MI455X (gfx1250) — compile-verified
//
#include <hip/hip_runtime.h>
#include <math.h>

#define NH 4
#define HD 256
#define HDIM 64
#define DEG 32
#define TOPK 10
#define NEG_SLOPE 0.2f

typedef __attribute__((ext_vector_type(2))) float v2f;
typedef __attribute__((ext_vector_type(8))) float v8f;

__device__ __forceinline__ float leaky(float x) { return x >= 0.f ? x : NEG_SLOPE * x; }

// ---------------------------------------------------------------------------
// C[M,256] = A[M,256] x B[256,256] using V_WMMA_F32_16X16X4_F32.
// Each block = 8 waves; each wave computes a 16x256 strip (16 N-tiles of 16).
// HEADS_B: B is W[4][256][64] interpreted as Bcat[k][h*64+e].
// TANH: fuse tanh epilogue (final layer).
// ---------------------------------------------------------------------------
template <bool HEADS_B, bool TANH>
__global__ __launch_bounds__(256) void gemm256_wmma(const float* __restrict__ A,
                                                    const float* __restrict__ B,
                                                    float* __restrict__ C, int M) {
  const int wave = threadIdx.x >> 5;
  const int lane = threadIdx.x & 31;
  const int half = lane >> 4;   // 0: lanes 0-15, 1: lanes 16-31
  const int l15  = lane & 15;
  const int m0 = blockIdx.x * 128 + wave * 16;
  if (m0 >= M) return;  // wave-uniform exit; EXEC stays all-ones for WMMA

  int arow = m0 + l15;
  if (arow > M - 1) arow = M - 1;  // clamp partial tile (stores are guarded)
  const float* ap = A + (size_t)arow * HD;

  v8f acc[16];
#pragma unroll
  for (int t = 0; t < 16; ++t) acc[t] = (v8f){0, 0, 0, 0, 0, 0, 0, 0};

  for (int k = 0; k < HD; k += 4) {
    const int kk = k + half * 2;
    v2f aF;
    aF.x = ap[kk];
    aF.y = ap[kk + 1];
#pragma unroll
    for (int t = 0; t < 16; ++t) {
      const int n = t * 16 + l15;
      v2f bF;
      if (HEADS_B) {
        const float* bp = B + (n >> 6) * (HD * HDIM) + kk * HDIM + (n & 63);
        bF.x = bp[0];
        bF.y = bp[HDIM];
      } else {
        const float* bp = B + kk * HD + n;
        bF.x = bp[0];
        bF.y = bp[HD];
      }
      acc[t] = __builtin_amdgcn_wmma_f32_16x16x4_f32(false, aF, false, bF,
                                                     (short)0, acc[t], false, false);
    }
  }

#pragma unroll
  for (int t = 0; t < 16; ++t) {
#pragma unroll
    for (int i = 0; i < 8; ++i) {
      const int r = m0 + half * 8 + i;  // C layout: vgpr i holds rows i / i+8
      if (r < M) {
        float v = acc[t][i];
        if (TANH) v = tanhf(v);
        C[(size_t)r * HD + t * 16 + l15] = v;
      }
    }
  }
}

// ---------------------------------------------------------------------------
// Per-row attention projections: out0[h][row] = dot(T[row, h*64:...], a[h][aOff0..])
// (and optionally out1 with aOff1). One wave per row; 8 lanes per head.
// ---------------------------------------------------------------------------
__global__ __launch_bounds__(256) void score_kernel(const float* __restrict__ T,
                                                    const float* __restrict__ a,
                                                    float* __restrict__ out0,
                                                    float* __restrict__ out1, int n,
                                                    int aOff0, int aOff1) {
  const int wave = threadIdx.x >> 5;
  const int lane = threadIdx.x & 31;
  const int row = blockIdx.x * 8 + wave;
  if (row >= n) return;
  const int h = lane >> 3;
  const int o = (lane & 7) * 8;
  const float* tp = T + (size_t)row * HD + h * HDIM + o;
  const float* a0 = a + h * (3 * HDIM) + aOff0 + o;
  const float* a1 = a + h * (3 * HDIM) + aOff1 + o;
  float s0 = 0.f, s1 = 0.f;
#pragma unroll
  for (int i = 0; i < 8; ++i) {
    const float t = tp[i];
    s0 += t * a0[i];
    if (out1) s1 += t * a1[i];
  }
#pragma unroll
  for (int off = 1; off < 8; off <<= 1) {
    s0 += __shfl_xor(s0, off, 32);
    s1 += __shfl_xor(s1, off, 32);
  }
  if ((lane & 7) == 0) {
    out0[(size_t)h * n + row] = s0;
    if (out1) out1[(size_t)h * n + row] = s1;
  }
}

// ---------------------------------------------------------------------------
// One wave per node: head-0 score over DEG=32 in-edges, stable top-10 (rank
// counting == lax.top_k tie-break), per-head edge softmax over selected lanes,
// weighted gather-accumulate of emb_t[src]+rel_t[rel] into neigh[node, 256].
// ---------------------------------------------------------------------------
__global__ __launch_bounds__(256) void aggregate_kernel(
    const int* __restrict__ src, const int* __restrict__ rid,
    const float* __restrict__ ss, const float* __restrict__ sd,
    const float* __restrict__ sr, const float* __restrict__ emb_t,
    const float* __restrict__ rel_t, float* __restrict__ neigh, int N, int R) {
  const int wave = threadIdx.x >> 5;
  const int lane = threadIdx.x & 31;
  const int nId = blockIdx.x * 8 + wave;
  if (nId >= N) return;

  const int s = src[(size_t)nId * DEG + lane];
  const int r = rid[(size_t)nId * DEG + lane];

  // head-0 score used for top-k in-edge selection
  const float sc0 = leaky(ss[s] + sd[nId] + sr[r]);

  int rank = 0;
#pragma unroll
  for (int j = 0; j < DEG; ++j) {
    const float sj = __shfl(sc0, j, 32);
    rank += (sj > sc0) || (sj == sc0 && j < lane);
  }
  const bool sel = rank < TOPK;
  const unsigned mask = (unsigned)__ballot(sel);

  float nrm[NH];
#pragma unroll
  for (int h = 0; h < NH; ++h) {
    const float sc =
        leaky(ss[(size_t)h * N + s] + sd[(size_t)h * N + nId] + sr[(size_t)h * R + r]);
    float v = sel ? sc : -1e30f;
#pragma unroll
    for (int o = 16; o > 0; o >>= 1) v = fmaxf(v, __shfl_xor(v, o, 32));
    const float ex = sel ? __expf(sc - v) : 0.f;
    float sm = ex;
#pragma unroll
    for (int o = 16; o > 0; o >>= 1) sm += __shfl_xor(sm, o, 32);
    nrm[h] = sel ? ex / sm : 0.f;
  }

  float acc[8] = {0, 0, 0, 0, 0, 0, 0, 0};
  for (int e = 0; e < DEG; ++e) {
    if (!((mask >> e) & 1u)) continue;
    const int se = __shfl(s, e, 32);
    const int re = __shfl(r, e, 32);
    float w[NH];
#pragma unroll
    for (int h = 0; h < NH; ++h) w[h] = __shfl(nrm[h], e, 32);
    const float* ep = emb_t + (size_t)se * HD;
    const float* rp = rel_t + (size_t)re * HD;
#pragma unroll
    for (int t = 0; t < 8; ++t) {
      const int j = lane + 32 * t;          // coalesced 128B per step
      acc[t] += w[t >> 1] * (ep[j] + rp[j]);  // head index (j>>6) == t>>1
    }
  }
#pragma unroll
  for (int t = 0; t < 8; ++t)
    neigh[(size_t)nId * HD + lane + 32 * t] = acc[t];
}

// ---------------------------------------------------------------------------
extern "C" void kernel_launch(void* const* d_in, const int* in_sizes, int n_in,
                              void* d_out, int out_size, void* d_ws, size_t ws_size,
                              hipStream_t stream) {
  const float* ent = (const float*)d_in[0];   // [N,256]
  const float* rel = (const float*)d_in[1];   // [R,256]
  const float* W   = (const float*)d_in[2];   // [4,256,64]
  const float* Wr  = (const float*)d_in[3];   // [4,256,64]
  const float* a   = (const float*)d_in[4];   // [4,192]
  const float* nw  = (const float*)d_in[5];   // [256,256]
  const int* src   = (const int*)d_in[6];     // [N,32]
  const int* rid   = (const int*)d_in[7];     // [N,32]
  const int N = in_sizes[0] / HD;
  const int R = in_sizes[1] / HD;

  // workspace layout (~105 MB total)
  char* ws = (char*)d_ws;
  size_t off = 0;
  auto up = [](size_t x) { return (x + 255) & ~(size_t)255; };
  float* emb_t = (float*)(ws + off); off = up(off + (size_t)N * HD * 4);
  float* rel_t = (float*)(ws + off); off = up(off + (size_t)R * HD * 4);
  float* ss    = (float*)(ws + off); off = up(off + (size_t)NH * N * 4);
  float* sd    = (float*)(ws + off); off = up(off + (size_t)NH * N * 4);
  float* sr    = (float*)(ws + off); off = up(off + (size_t)NH * R * 4);
  float* neigh = (float*)(ws + off); off = up(off + (size_t)N * HD * 4);
  (void)ws_size; (void)n_in; (void)out_size;

  const dim3 blk(256);
  // 1) emb_t = ent_emb @ Wcat   (WMMA f32)
  gemm256_wmma<true, false><<<(N + 127) / 128, blk, 0, stream>>>(ent, W, emb_t, N);
  // 2) rel_t = rel_emb @ Wrcat  (WMMA f32)
  gemm256_wmma<true, false><<<(R + 127) / 128, blk, 0, stream>>>(rel, Wr, rel_t, R);
  // 3) ss/sd per node, sr per relation
  score_kernel<<<(N + 7) / 8, blk, 0, stream>>>(emb_t, a, ss, sd, N, 0, HDIM);
  score_kernel<<<(R + 7) / 8, blk, 0, stream>>>(rel_t, a, sr, nullptr, R, 2 * HDIM, 2 * HDIM);
  // 4) top-k + edge softmax + weighted aggregate
  aggregate_kernel<<<(N + 7) / 8, blk, 0, stream>>>(src, rid, ss, sd, sr, emb_t,
                                                    rel_t, neigh, N, R);
  // 5) out = tanh(neigh @ neigh_w)   (WMMA f32, fused tanh)
  gemm256_wmma<false, true><<<(N + 127) / 128, blk, 0, stream>>>(neigh, nw,
                                                                 (float*)d_out, N);
}